// FusionHydraPredictor_63866163692303
// MI455X (gfx1250) — compile-verified
//
#include <hip/hip_runtime.h>

typedef __attribute__((ext_vector_type(2))) float v2f;
typedef __attribute__((ext_vector_type(8))) float v8f;

#define B_      4096
#define DIN     512
#define DH      128
#define KHEADS  16
#define M_      1250
#define NGENES  20000

#define ROWS        64    // batch rows per block
#define THREADS     128   // 4 wave32s
#define ZS_STRIDE   68    // 64 cols + pad; 272B rows -> 16B aligned for b128
#define P1STR       288   // W1 pair-row stride (floats): 256 + 32 -> lo/hi rows in disjoint bank halves
#define P2STR       32    // W2 pair-row stride (floats): 32 mod 64 -> disjoint halves, no padding
#define HS_STRIDE   130   // h row stride (even -> 8B-aligned b64 A-frag reads)

static __device__ __forceinline__ v8f wmma_f32(v2f a, v2f b, v8f c) {
  // D = A(16x4,f32) x B(4x16,f32) + C(16x16,f32) -> v_wmma_f32_16x16x4_f32
  return __builtin_amdgcn_wmma_f32_16x16x4_f32(false, a, false, b, (short)0, c,
                                               false, false);
}

// Async global->LDS copies (ASYNCcnt-tracked DMA path, CDNA5).
// Generic pointers to LDS have the wave-relative LDS offset in their low 32 bits.
static __device__ __forceinline__ void async_copy_b128(void* lds_dst, const void* gsrc) {
  unsigned l = (unsigned)(unsigned long long)lds_dst;
  asm volatile("global_load_async_to_lds_b128 %0, %1, off"
               :: "v"(l), "v"(gsrc) : "memory");
}
static __device__ __forceinline__ void async_copy_b32(void* lds_dst, const void* gsrc) {
  unsigned l = (unsigned)(unsigned long long)lds_dst;
  asm volatile("global_load_async_to_lds_b32 %0, %1, off"
               :: "v"(l), "v"(gsrc) : "memory");
}
static __device__ __forceinline__ void wait_async_zero() {
  asm volatile("s_wait_asynccnt 0" ::: "memory");
}

__global__ __launch_bounds__(THREADS)
void hydra_fused(const float* __restrict__ z,  const float* __restrict__ W1,
                 const float* __restrict__ b1, const float* __restrict__ W2,
                 const float* __restrict__ b2, const int* __restrict__ idx,
                 float* __restrict__ full, float* __restrict__ per_head)
{
  __shared__ union {
    struct { float zs[ROWS * ZS_STRIDE]; float w1p[32 * P1STR]; } p1; // 54.3 KB
    struct { float hs[ROWS * HS_STRIDE]; float w2p[64 * P2STR]; } p2; // 41.5 KB
  } sm;

  const int t    = threadIdx.x;
  const int wave = t >> 5;
  const int lane = t & 31;
  const int lm   = lane & 15;     // M/N position within a 16-wide tile
  const int hi   = lane >> 4;     // high half-wave flag
  const int rW   = wave * 16;     // this wave's row stripe within the block tile
  const int rowBase = blockIdx.x * ROWS;
  const int k       = blockIdx.y;

  // ---------------- Phase 1: h = relu(z @ W1[k] + b1[k]) ----------------
  v8f accH[8];
  #pragma unroll
  for (int c = 0; c < 8; ++c) accH[c] = v8f{};

  for (int kk0 = 0; kk0 < DIN; kk0 += 64) {
    // Async-stage z tile: ROWS x 64, contiguous 16B chunks.
    #pragma unroll
    for (int i = 0; i < 8; ++i) {
      int q = t + i * THREADS;            // 0..1023
      int r = q >> 4, c4 = q & 15;
      async_copy_b128(&sm.p1.zs[r * ZS_STRIDE + c4 * 4],
                      &z[(size_t)(rowBase + r) * DIN + kk0 + c4 * 4]);
    }
    // Async-stage W1 tile into pair-interleaved layout:
    // w1p[(d/2)*P1STR + col*2 + (d&1)] = W1[k][kk0+d][col].
    // Thread t owns column col = t, rows d = 0..63.
    #pragma unroll
    for (int d = 0; d < 64; ++d) {
      async_copy_b32(&sm.p1.w1p[(d >> 1) * P1STR + t * 2 + (d & 1)],
                     &W1[((size_t)(k * DIN + kk0 + d)) * DH + t]);
    }
    wait_async_zero();
    __syncthreads();

    for (int kk = 0; kk < 64; kk += 4) {
      // A frag: row = rW+lm, depth = kk + 2*hi .. +1  (single b64)
      v2f a = *(const v2f*)&sm.p1.zs[(rW + lm) * ZS_STRIDE + kk + 2 * hi];
      #pragma unroll
      for (int c = 0; c < 8; ++c) {
        // B frag: rows kk+2hi, kk+2hi+1 at col c*16+lm  (single b64, pair layout)
        v2f b = *(const v2f*)&sm.p1.w1p[((kk >> 1) + hi) * P1STR + (c * 16 + lm) * 2];
        accH[c] = wmma_f32(a, b, accH[c]);
      }
    }
    __syncthreads();
  }

  // Bias + ReLU; park h tile in LDS (C-layout: VGPR i -> row i + 8*hi, col lm)
  #pragma unroll
  for (int c = 0; c < 8; ++c) {
    float bv = b1[k * DH + c * 16 + lm];
    #pragma unroll
    for (int i = 0; i < 8; ++i) {
      float v = accH[c][i] + bv;
      v = v > 0.0f ? v : 0.0f;
      sm.p2.hs[(rW + i + 8 * hi) * HS_STRIDE + c * 16 + lm] = v;
    }
  }
  __syncthreads();

  // Preload all 32 A-fragments of h for this wave's 16 rows (stays in VGPRs)
  v2f af[32];
  #pragma unroll
  for (int j = 0; j < 32; ++j)
    af[j] = *(const v2f*)&sm.p2.hs[(rW + lm) * HS_STRIDE + 4 * j + 2 * hi];

  // ---------------- Phase 2: out = h @ W2[k] + b2[k]; scatter ----------------
  const int NT = (M_ + 15) / 16;          // 79 column tiles
  for (int ct = 0; ct < NT; ++ct) {
    __syncthreads();                      // previous tile's w2p reads are done
    // Async-stage W2 slab (128 rows x 16 cols) into pair layout.
    // Column clamp for the partial last tile: clamped garbage only reaches
    // accumulator lanes whose column >= M_, which are never stored.
    #pragma unroll
    for (int i = 0; i < 16; ++i) {
      int q = t + i * THREADS;            // 0..2047
      int d = q >> 4, c = q & 15;
      int col  = ct * 16 + c;
      int colc = col < M_ ? col : (M_ - 1);
      async_copy_b32(&sm.p2.w2p[(d >> 1) * P2STR + c * 2 + (d & 1)],
                     &W2[((size_t)(k * DH + d)) * M_ + colc]);
    }
    wait_async_zero();
    __syncthreads();

    v8f acc = v8f{};
    #pragma unroll
    for (int j = 0; j < 32; ++j) {
      // B frag: rows 4j+2hi, 4j+2hi+1 at col lm (single b64, pair layout)
      v2f b = *(const v2f*)&sm.p2.w2p[(2 * j + hi) * P2STR + lm * 2];
      acc = wmma_f32(af[j], b, acc);
    }

    int col = ct * 16 + lm;
    if (col < M_) {
      float b2v = b2[k * M_ + col];
      int g = idx[k * M_ + col];          // scatter column in canonical order
      #pragma unroll
      for (int i = 0; i < 8; ++i) {
        int row = rowBase + rW + i + 8 * hi;
        float v = acc[i] + b2v;
        full[(size_t)row * NGENES + g] = v;
        per_head[((size_t)k * B_ + row) * M_ + col] = v;
      }
    }
  }
}

extern "C" void kernel_launch(void* const* d_in, const int* in_sizes, int n_in,
                              void* d_out, int out_size, void* d_ws, size_t ws_size,
                              hipStream_t stream) {
  const float* z   = (const float*)d_in[0];
  const float* W1  = (const float*)d_in[1];
  const float* b1  = (const float*)d_in[2];
  const float* W2  = (const float*)d_in[3];
  const float* b2  = (const float*)d_in[4];
  const int*   idx = (const int*)d_in[5];

  float* full     = (float*)d_out;
  float* per_head = full + (size_t)B_ * NGENES;

  dim3 grid(B_ / ROWS, KHEADS);
  hipLaunchKernelGGL(hydra_fused, grid, dim3(THREADS), 0, stream,
                     z, W1, b1, W2, b2, idx, full, per_head);
}